// GNNEncoder_10471130268372
// MI455X (gfx1250) — compile-verified
//
#include <hip/hip_runtime.h>
#include <hip/hip_bf16.h>
#include <math.h>

typedef float v2f __attribute__((ext_vector_type(2)));
typedef float v8f __attribute__((ext_vector_type(8)));

#define CC 128   // channels
#define HH 4     // heads
#define DD 32    // head dim

// ---------------------------------------------------------------- utilities

__device__ __forceinline__ float gelu_f(float x) {
  // tanh-approx gelu (jax.nn.gelu default approximate=True)
  float x3 = x * x * x;
  return 0.5f * x * (1.0f + tanhf(0.7978845608028654f * (x + 0.044715f * x3)));
}

// monotone float<->uint mapping so atomicMax(u32) implements float max
__device__ __forceinline__ unsigned encF(float f) {
  unsigned u = __float_as_uint(f);
  return (u & 0x80000000u) ? ~u : (u | 0x80000000u);
}
__device__ __forceinline__ float decF(unsigned u) {
  return (u & 0x80000000u) ? __uint_as_float(u & 0x7fffffffu) : __uint_as_float(~u);
}
#define ENC_NEG_INF 0x007FFFFFu  // encF(-inf)

__global__ void fill_f32_k(float* p, float v, int n) {
  int i = blockIdx.x * blockDim.x + threadIdx.x;
  if (i < n) p[i] = v;
}
__global__ void fill_u32_k(unsigned* p, unsigned v, int n) {
  int i = blockIdx.x * blockDim.x + threadIdx.x;
  if (i < n) p[i] = v;
}

// ---------------------------------------------------------------- f32 WMMA GEMM
// out[M x 128] = op(A[M x K]) @ W[K x 128] + bias, epilogue per MODE.
// MODE 0: relu(out)                      (lin0)
// MODE 1: out                            (K/Q/V projections)
// MODE 2: A pre-activated with gelu; out = sg*out + (1-sg)*resid, sg=sigmoid(skip)
// One wave computes a 32-row x 128-col tile: two 16-row WMMA M-tiles share every
// B fragment (8 v_wmma per 8 B-element loads per M-tile pair -> 1.25 loads/wmma).
template <int MODE>
__global__ __launch_bounds__(128) void gemm128_wmma(
    const float* __restrict__ A, int K, const float* __restrict__ W,
    const float* __restrict__ bias, const float* __restrict__ resid,
    const float* __restrict__ skip, float* __restrict__ out, int M) {
  const int lane = threadIdx.x & 31;
  const int wv = threadIdx.x >> 5;
  const int m0 = (blockIdx.x * 4 + wv) * 32;
  if (m0 >= M) return;  // whole-wave uniform: EXEC stays all-ones for WMMA
  const int half = lane >> 4;  // 0: K=0,1 / rows 0-7 ; 1: K=2,3 / rows 8-15
  const int l = lane & 15;
  const int kh = 2 * half;

  v8f acc[2][8];
#pragma unroll
  for (int mt = 0; mt < 2; ++mt)
#pragma unroll
    for (int t = 0; t < 8; ++t)
#pragma unroll
      for (int j = 0; j < 8; ++j) acc[mt][t][j] = 0.0f;

  // clamped duplicate rows are fine: their stores are masked below
  int r0 = m0 + l;       if (r0 >= M) r0 = M - 1;
  int r1 = m0 + 16 + l;  if (r1 >= M) r1 = M - 1;
  const float* Arow0 = A + (size_t)r0 * K;
  const float* Arow1 = A + (size_t)r1 * K;

  for (int k = 0; k < K; k += 4) {
    v2f a0 = *(const v2f*)(Arow0 + k + kh);
    v2f a1 = *(const v2f*)(Arow1 + k + kh);
    if (MODE == 2) {
      a0.x = gelu_f(a0.x); a0.y = gelu_f(a0.y);
      a1.x = gelu_f(a1.x); a1.y = gelu_f(a1.y);
    }
    v2f b[8];
#pragma unroll
    for (int t = 0; t < 8; ++t) {
      const int n = t * 16 + l;
      b[t].x = W[(size_t)(k + kh) * CC + n];
      b[t].y = W[(size_t)(k + kh + 1) * CC + n];
    }
#pragma unroll
    for (int t = 0; t < 8; ++t)
      acc[0][t] = __builtin_amdgcn_wmma_f32_16x16x4_f32(false, a0, false, b[t],
                                                        (short)0, acc[0][t], false, false);
#pragma unroll
    for (int t = 0; t < 8; ++t)
      acc[1][t] = __builtin_amdgcn_wmma_f32_16x16x4_f32(false, a1, false, b[t],
                                                        (short)0, acc[1][t], false, false);
  }

  float sg = 0.0f, omsg = 0.0f;
  if (MODE == 2) { sg = 1.0f / (1.0f + __expf(-skip[0])); omsg = 1.0f - sg; }

#pragma unroll
  for (int mt = 0; mt < 2; ++mt) {
#pragma unroll
    for (int t = 0; t < 8; ++t) {
      const int col = t * 16 + l;
      const float bcol = bias[col];
#pragma unroll
      for (int rr = 0; rr < 8; ++rr) {
        const int row = m0 + mt * 16 + rr + 8 * half;
        if (row < M) {
          float dv = acc[mt][t][rr] + bcol;
          if (MODE == 0) dv = fmaxf(dv, 0.0f);
          if (MODE == 2) dv = sg * dv + omsg * resid[(size_t)row * CC + col];
          out[(size_t)row * CC + col] = dv;
        }
      }
    }
  }
}

// ---------------------------------------------------------------- edge kernels
// alpha[e,h] = (q[t,h,:] . (k[s,h,:] @ a_rel[h])) * p_rel[h] / sqrt(D)
__global__ __launch_bounds__(256) void edge_alpha_k(
    const float* __restrict__ kbuf, const float* __restrict__ qbuf,
    const int* __restrict__ ei, int E, const float* __restrict__ a_rel,
    const float* __restrict__ p_rel, float* __restrict__ alpha) {
  int idx = blockIdx.x * blockDim.x + threadIdx.x;
  if (idx >= E * HH) return;
  int e = idx >> 2, h = idx & 3;
  int s = ei[e], t = ei[E + e];
  const float* kv = kbuf + (size_t)s * CC + h * DD;
  const float* qv = qbuf + (size_t)t * CC + h * DD;
  const float* Ar = a_rel + h * DD * DD;
  float krel[DD];
#pragma unroll
  for (int f = 0; f < DD; ++f) krel[f] = 0.0f;
  for (int d = 0; d < DD; ++d) {
    float kd = kv[d];
#pragma unroll
    for (int f = 0; f < DD; ++f) krel[f] = fmaf(kd, Ar[d * DD + f], krel[f]);
  }
  float acc = 0.0f;
#pragma unroll
  for (int f = 0; f < DD; ++f) acc = fmaf(qv[f], krel[f], acc);
  alpha[idx] = acc * p_rel[h] * 0.17677669529663687f;  // 1/sqrt(32)
}

__global__ __launch_bounds__(256) void edge_amax_k(
    const float* __restrict__ alpha, const int* __restrict__ ei, int E,
    unsigned* __restrict__ amax) {
  int idx = blockIdx.x * blockDim.x + threadIdx.x;
  if (idx >= E * HH) return;
  int e = idx >> 2, h = idx & 3;
  int t = ei[E + e];
  atomicMax(&amax[(size_t)t * HH + h], encF(alpha[idx]));
}

__global__ __launch_bounds__(256) void edge_expden_k(
    float* __restrict__ alpha, const int* __restrict__ ei, int E,
    const unsigned* __restrict__ amax, float* __restrict__ den) {
  int idx = blockIdx.x * blockDim.x + threadIdx.x;
  if (idx >= E * HH) return;
  int e = idx >> 2, h = idx & 3;
  int t = ei[E + e];
  float am = decF(amax[(size_t)t * HH + h]);
  float ex = __expf(alpha[idx] - am);
  alpha[idx] = ex;  // reuse buffer: holds exp from here on
  __hip_atomic_fetch_add(&den[(size_t)t * HH + h], ex, __ATOMIC_RELAXED,
                         __HIP_MEMORY_SCOPE_AGENT);
}

// agg[t,h,:] += softmax(e) * (v[s,h,:] @ m_rel[h])
__global__ __launch_bounds__(256) void edge_msg_k(
    const float* __restrict__ vbuf, const float* __restrict__ alpha,
    const int* __restrict__ ei, int E, const float* __restrict__ m_rel,
    const float* __restrict__ den, float* __restrict__ agg) {
  int idx = blockIdx.x * blockDim.x + threadIdx.x;
  if (idx >= E * HH) return;
  int e = idx >> 2, h = idx & 3;
  int s = ei[e], t = ei[E + e];
  float a = alpha[idx] / (den[(size_t)t * HH + h] + 1e-16f);
  const float* vv = vbuf + (size_t)s * CC + h * DD;
  const float* Mr = m_rel + h * DD * DD;
  float mres[DD];
#pragma unroll
  for (int f = 0; f < DD; ++f) mres[f] = 0.0f;
  for (int d = 0; d < DD; ++d) {
    float vd = vv[d];
#pragma unroll
    for (int f = 0; f < DD; ++f) mres[f] = fmaf(vd, Mr[d * DD + f], mres[f]);
  }
  float* ag = agg + (size_t)t * CC + h * DD;
#pragma unroll
  for (int f = 0; f < DD; ++f)
    __hip_atomic_fetch_add(&ag[f], a * mres[f], __ATOMIC_RELAXED,
                           __HIP_MEMORY_SCOPE_AGENT);
}

// ---------------------------------------------------------------- host launch

extern "C" void kernel_launch(void* const* d_in, const int* in_sizes, int n_in,
                              void* d_out, int out_size, void* d_ws, size_t ws_size,
                              hipStream_t stream) {
  (void)n_in; (void)out_size; (void)ws_size;
  const int FA = 64, FP = 128;
  const float* x_author = (const float*)d_in[0];
  const float* x_paper  = (const float*)d_in[1];
  const int* ei_writes = (const int*)d_in[2];
  const int* ei_rev    = (const int*)d_in[3];
  const int* ei_cites  = (const int*)d_in[4];
  const int NA = in_sizes[0] / FA;
  const int NP = in_sizes[1] / FP;
  const int EW = in_sizes[2] / 2, ER = in_sizes[3] / 2, ECI = in_sizes[4] / 2;
  int Emax = EW; if (ER > Emax) Emax = ER; if (ECI > Emax) Emax = ECI;
  int Nmax = (NA > NP) ? NA : NP;

  auto P = [&](int i) { return (const float*)d_in[i]; };
  // params (JAX pytree order): layers[0], layers[1], lin0
  //   layer base = 5 + 27*l:
  //     +0..2  edge.cites      (a_rel, m_rel, p_rel)
  //     +3..5  edge.rev_writes (a_rel, m_rel, p_rel)
  //     +6..8  edge.writes     (a_rel, m_rel, p_rel)
  //     +9..17 node.author (Wa,Wk,Wq,Wv,ba,bk,bq,bv,skip)
  //     +18..26 node.paper (same order)
  //   lin0: 59=author.W, 60=author.b, 61=paper.W, 62=paper.b

  // workspace layout (floats)
  float* w = (float*)d_ws;
  const size_t oA = (size_t)NA * CC, oP = (size_t)NP * CC;
  float* xa0 = w;            float* xa1 = xa0 + oA;
  float* xp0 = xa1 + oA;     float* xp1 = xp0 + oP;
  float* ka = xp1 + oP;      float* qa = ka + oA;  float* va = qa + oA;
  float* kp = va + oA;       float* qp = kp + oP;  float* vp = qp + oP;
  float* agg_a = vp + oP;    float* agg_p = agg_a + oA;
  float* alpha = agg_p + oP;                       // Emax*HH
  unsigned* amax = (unsigned*)(alpha + (size_t)Emax * HH);  // Nmax*HH
  float* den = (float*)(amax + (size_t)Nmax * HH);          // Nmax*HH

  auto gemm = [&](int mode, const float* A, int K, const float* W,
                  const float* b, const float* resid, const float* skip,
                  float* out, int M) {
    dim3 grid((M + 127) / 128), block(128);
    if (mode == 0)
      gemm128_wmma<0><<<grid, block, 0, stream>>>(A, K, W, b, resid, skip, out, M);
    else if (mode == 1)
      gemm128_wmma<1><<<grid, block, 0, stream>>>(A, K, W, b, resid, skip, out, M);
    else
      gemm128_wmma<2><<<grid, block, 0, stream>>>(A, K, W, b, resid, skip, out, M);
  };
  auto fillf = [&](float* p, float v, int n) {
    fill_f32_k<<<(n + 255) / 256, 256, 0, stream>>>(p, v, n);
  };
  auto fillu = [&](unsigned* p, unsigned v, int n) {
    fill_u32_k<<<(n + 255) / 256, 256, 0, stream>>>(p, v, n);
  };

  // lin0 + relu
  gemm(0, x_author, FA, P(59), P(60), nullptr, nullptr, xa0, NA);
  gemm(0, x_paper,  FP, P(61), P(62), nullptr, nullptr, xp0, NP);

  float* outA = (float*)d_out;
  float* outP = (float*)d_out + oA;

  const float* xa_cur = xa0;
  const float* xp_cur = xp0;
  for (int l = 0; l < 2; ++l) {
    const int base = 5 + 27 * l;
    // K/Q/V projections
    gemm(1, xa_cur, CC, P(base + 10), P(base + 14), nullptr, nullptr, ka, NA);
    gemm(1, xa_cur, CC, P(base + 11), P(base + 15), nullptr, nullptr, qa, NA);
    gemm(1, xa_cur, CC, P(base + 12), P(base + 16), nullptr, nullptr, va, NA);
    gemm(1, xp_cur, CC, P(base + 19), P(base + 23), nullptr, nullptr, kp, NP);
    gemm(1, xp_cur, CC, P(base + 20), P(base + 24), nullptr, nullptr, qp, NP);
    gemm(1, xp_cur, CC, P(base + 21), P(base + 25), nullptr, nullptr, vp, NP);

    fillf(agg_a, 0.0f, (int)oA);
    fillf(agg_p, 0.0f, (int)oP);

    auto do_rel = [&](const int* ei, int E, const float* kb, const float* vb,
                      const float* qb, float* aggd, int Ndst,
                      const float* arel, const float* mrel, const float* prel) {
      const int nT = E * HH, gb = (nT + 255) / 256;
      fillu(amax, ENC_NEG_INF, Ndst * HH);
      fillf(den, 0.0f, Ndst * HH);
      edge_alpha_k<<<gb, 256, 0, stream>>>(kb, qb, ei, E, arel, prel, alpha);
      edge_amax_k<<<gb, 256, 0, stream>>>(alpha, ei, E, amax);
      edge_expden_k<<<gb, 256, 0, stream>>>(alpha, ei, E, amax, den);
      edge_msg_k<<<gb, 256, 0, stream>>>(vb, alpha, ei, E, mrel, den, aggd);
    };
    // writes: author -> paper
    do_rel(ei_writes, EW, ka, va, qp, agg_p, NP, P(base + 6), P(base + 7), P(base + 8));
    // rev_writes: paper -> author
    do_rel(ei_rev, ER, kp, vp, qa, agg_a, NA, P(base + 3), P(base + 4), P(base + 5));
    // cites: paper -> paper
    do_rel(ei_cites, ECI, kp, vp, qp, agg_p, NP, P(base + 0), P(base + 1), P(base + 2));

    // output projection: gelu(agg) @ Wa + ba, skip blend
    float* oa = (l == 1) ? outA : xa1;
    float* op = (l == 1) ? outP : xp1;
    gemm(2, agg_a, CC, P(base + 9), P(base + 13), xa_cur, P(base + 17), oa, NA);
    gemm(2, agg_p, CC, P(base + 18), P(base + 22), xp_cur, P(base + 26), op, NP);
    xa_cur = oa;
    xp_cur = op;
  }
}